// GraphConv_32504312496863
// MI455X (gfx1250) — compile-verified
//
#include <hip/hip_runtime.h>
#include <stdint.h>

#define EMB_DIM 64
#define TILE    1024          // edges staged per TDM tile
#define BLOCK   256           // 8 waves of 32
#define NWAVES  (BLOCK / 32)

typedef unsigned int u32x4 __attribute__((ext_vector_type(4)));
typedef int          i32x4 __attribute__((ext_vector_type(4)));
typedef int          i32x8 __attribute__((ext_vector_type(8)));

#if __has_include(<hip/amd_detail/amd_gfx1250_TDM.h>)
#define TDM_ARITY6 1
#endif

__device__ __forceinline__ uint32_t lds_off(const void* p) {
  // Generic pointers to LDS carry the LDS byte offset in the low 32 bits
  // (flat-aperture scheme).
  return (uint32_t)(uintptr_t)p;
}

// 1D TDM descriptor: copy nelem 4-byte elements from gsrc to LDS at lds_addr.
// D# layout per CDNA5 ISA ch. 8 (group0: count/lds_addr/global_addr/type=2,
// group1: data_size=4B, tensor_dim0=tile_dim0=nelem, dims 1/2 collapsed to 1).
__device__ __forceinline__ void tdm_load_1d(const void* gsrc, uint32_t lds_addr,
                                            uint32_t nelem) {
  uint64_t ga = (uint64_t)(uintptr_t)gsrc;
  u32x4 g0;
  g0.x = 1u;                                                // count=1, user desc
  g0.y = lds_addr;                                          // lds_addr[31:0]
  g0.z = (uint32_t)ga;                                      // global_addr[31:0]
  g0.w = (uint32_t)((ga >> 32) & 0x01FFFFFFu) | (2u << 30); // ga[56:32] | type=2
  i32x8 g1;
  g1[0] = (int)(2u << 16);                        // data_size = 2 -> 4 bytes
  g1[1] = (int)((nelem & 0xFFFFu) << 16);         // tensor_dim0[15:0]
  g1[2] = (int)((nelem >> 16) & 0xFFFFu)          // tensor_dim0[31:16]
        | (int)(1u << 16);                        // tensor_dim1 = 1
  g1[3] = (int)((nelem & 0xFFFFu) << 16);         // tile_dim0 = nelem (<=64K)
  g1[4] = (int)(1u | (1u << 16));                 // tile_dim1 = 1, tile_dim2 = 1
  g1[5] = (int)nelem;                             // tensor_dim0_stride[31:0]
  g1[6] = 0;                                      // stride hi / dim1_stride lo
  g1[7] = 0;
  i32x4 z4 = {0, 0, 0, 0};
#ifdef TDM_ARITY6
  i32x8 z8 = {0, 0, 0, 0, 0, 0, 0, 0};
  __builtin_amdgcn_tensor_load_to_lds(g0, g1, z4, z4, z8, 0);
#else
  __builtin_amdgcn_tensor_load_to_lds(g0, g1, z4, z4, 0);
#endif
}

// COO SpMM scatter: one wave per edge, 2 dims per lane (64 = 32 lanes x 2).
// Edge tiles are double-buffered in LDS via the Tensor Data Mover; wave 0
// issues the DMA for tile i+1, waits tensorcnt<=3 (in-order => tile i done),
// then the block gathers src rows (L2-resident) and scatters with
// non-returning global_atomic_add_f32.
__global__ __launch_bounds__(BLOCK) void spmm_edges_kernel(
    const int* __restrict__ rows, const int* __restrict__ cols,
    const float* __restrict__ vals, const float* __restrict__ src,
    float* __restrict__ dst, int n_edges) {
  __shared__ int   s_rows[2][TILE];
  __shared__ int   s_cols[2][TILE];
  __shared__ float s_vals[2][TILE];

  const int lane = threadIdx.x & 31;
  const int wave = threadIdx.x >> 5;
  const long long stride = (long long)gridDim.x * TILE;
  long long base = (long long)blockIdx.x * TILE;
  if (base >= n_edges) return;

  int buf = 0;
  {
    long long rem = (long long)n_edges - base;
    uint32_t cnt0 = (uint32_t)(rem < TILE ? rem : TILE);
    if (wave == 0) {
      tdm_load_1d(rows + base, lds_off(&s_rows[0][0]), cnt0);
      tdm_load_1d(cols + base, lds_off(&s_cols[0][0]), cnt0);
      tdm_load_1d(vals + base, lds_off(&s_vals[0][0]), cnt0);
    }
  }

  for (; base < n_edges; base += stride) {
    long long rem = (long long)n_edges - base;
    int cnt = (int)(rem < TILE ? rem : TILE);
    long long nbase = base + stride;

    if (wave == 0) {
      if (nbase < n_edges) {
        long long nrem = (long long)n_edges - nbase;
        uint32_t ncnt = (uint32_t)(nrem < TILE ? nrem : TILE);
        int nb = buf ^ 1;
        tdm_load_1d(rows + nbase, lds_off(&s_rows[nb][0]), ncnt);
        tdm_load_1d(cols + nbase, lds_off(&s_cols[nb][0]), ncnt);
        tdm_load_1d(vals + nbase, lds_off(&s_vals[nb][0]), ncnt);
        __builtin_amdgcn_s_wait_tensorcnt(3);  // current tile's 3 ops done
      } else {
        __builtin_amdgcn_s_wait_tensorcnt(0);
      }
    }
    __syncthreads();  // LDS tile visible to all waves

    for (int e = wave; e < cnt; e += NWAVES) {
      int   r = s_rows[buf][e];
      int   c = s_cols[buf][e];
      float v = s_vals[buf][e];
      // 256B coalesced gather of one embedding row per wave (hits L2).
      const float2 g =
          *(reinterpret_cast<const float2*>(src + (size_t)c * EMB_DIM) + lane);
      float* dp = dst + (size_t)r * EMB_DIM + 2 * lane;
      atomicAdd(dp + 0, v * g.x);   // global_atomic_add_f32, no return
      atomicAdd(dp + 1, v * g.y);
    }
    __syncthreads();  // tile fully consumed before its buffer is re-filled
    buf ^= 1;
  }
}

__global__ void zero3_kernel(float4* __restrict__ a, float4* __restrict__ b,
                             float4* __restrict__ c, int n4) {
  int i = blockIdx.x * blockDim.x + threadIdx.x;
  if (i < n4) {
    float4 z = {0.f, 0.f, 0.f, 0.f};
    a[i] = z; b[i] = z; c[i] = z;
  }
}

// out currently holds layer-3; out = 0.25 * (e0 + t0 + t1 + out)
__global__ void finalize_kernel(float4* __restrict__ out,
                                const float4* __restrict__ e0,
                                const float4* __restrict__ t0,
                                const float4* __restrict__ t1, int n4) {
  int i = blockIdx.x * blockDim.x + threadIdx.x;
  if (i < n4) {
    float4 a = e0[i], b = t0[i], c = t1[i], d = out[i];
    float4 r;
    r.x = 0.25f * (a.x + b.x + c.x + d.x);
    r.y = 0.25f * (a.y + b.y + c.y + d.y);
    r.z = 0.25f * (a.z + b.z + c.z + d.z);
    r.w = 0.25f * (a.w + b.w + c.w + d.w);
    out[i] = r;
  }
}

extern "C" void kernel_launch(void* const* d_in, const int* in_sizes, int n_in,
                              void* d_out, int out_size, void* d_ws,
                              size_t ws_size, hipStream_t stream) {
  const float* emb  = (const float*)d_in[0];
  const int*   rows = (const int*)d_in[1];
  const int*   cols = (const int*)d_in[2];
  const float* vals = (const float*)d_in[3];
  const int n_nodes = in_sizes[0] / EMB_DIM;  // 150000
  const int n_edges = in_sizes[1];            // 4800000

  float* t0  = (float*)d_ws;
  float* t1  = t0 + (size_t)n_nodes * EMB_DIM;
  float* out = (float*)d_out;

  const int nflt = n_nodes * EMB_DIM;
  const int n4   = nflt / 4;
  const int eb   = (n4 + BLOCK - 1) / BLOCK;

  zero3_kernel<<<eb, BLOCK, 0, stream>>>((float4*)t0, (float4*)t1, (float4*)out,
                                         n4);

  int ntiles = (n_edges + TILE - 1) / TILE;
  int grid = ntiles < 2048 ? ntiles : 2048;  // persistent blocks -> TDM overlap

  spmm_edges_kernel<<<grid, BLOCK, 0, stream>>>(rows, cols, vals, emb, t0,
                                                n_edges);
  spmm_edges_kernel<<<grid, BLOCK, 0, stream>>>(rows, cols, vals, t0, t1,
                                                n_edges);
  spmm_edges_kernel<<<grid, BLOCK, 0, stream>>>(rows, cols, vals, t1, out,
                                                n_edges);

  finalize_kernel<<<eb, BLOCK, 0, stream>>>((float4*)out, (const float4*)emb,
                                            (const float4*)t0,
                                            (const float4*)t1, n4);
}